// NearestNeighborLoss_2070174237072
// MI455X (gfx1250) — compile-verified
//
#include <hip/hip_runtime.h>

typedef float v2f __attribute__((ext_vector_type(2)));
typedef float v4f __attribute__((ext_vector_type(4)));
typedef float v8f __attribute__((ext_vector_type(8)));

// Kernel 1: stream both tensors once (float4 NT loads), accumulate
// w(r)*w(s)*(f-t)^2 per thread, then reduce:
//   thread -> wave   : V_WMMA_F32_16X16X4_F32 with B = ones (matrix-unit reduce)
//   wave   -> block  : LDS
//   block  -> d_ws   : one float partial per block
__global__ void __launch_bounds__(256) nnloss_partial(
    const float* __restrict__ fake, const float* __restrict__ tgt,
    float* __restrict__ partials, int nvec4)
{
    constexpr int W = 256, H = 256;
    const v4f* __restrict__ f4 = (const v4f*)fake;
    const v4f* __restrict__ t4 = (const v4f*)tgt;

    float acc = 0.0f;
    const int stride = gridDim.x * blockDim.x;
    for (int v = blockIdx.x * blockDim.x + threadIdx.x; v < nvec4; v += stride) {
        v4f f = __builtin_nontemporal_load(f4 + v);  // global_load_b128, TH=NT
        v4f t = __builtin_nontemporal_load(t4 + v);

        const int e = v << 2;               // flat element index of lane 0 of this vec4
        const int s = e & (W - 1);          // column of first element (multiple of 4)
        const int r = (e >> 8) & (H - 1);   // row

        // Separable reflect+coverage weights: [2,4,3,...,3,4,2] per axis.
        const float wr = 3.0f + (float)((r == 1) + (r == H - 2) - (r == 0) - (r == H - 1));
        const float w0 = 3.0f - (float)(s == 0);       // cols {0,1}: 2,4
        const float w1 = 3.0f + (float)(s == 0);
        const float w2 = 3.0f + (float)(s == W - 4);   // cols {254,255}: 4,2
        const float w3 = 3.0f - (float)(s == W - 4);

        const float d0 = f[0] - t[0], d1 = f[1] - t[1];
        const float d2 = f[2] - t[2], d3 = f[3] - t[3];
        const float ssum = fmaf(w0, d0 * d0,
                           fmaf(w1, d1 * d1,
                           fmaf(w2, d2 * d2, w3 * (d3 * d3))));
        acc = fmaf(wr, ssum, acc);
    }

    // Wave32 reduction on the matrix unit: D = A x ones + 0.
    // A (16x4 f32) holds one partial per lane in VGPR0 (VGPR1 = 0); with B all
    // ones, every D[m][n] = rowsum(m). Summing a lane's 8 D regs gives
    // sum(rows 0-7) on lanes 0-15 and sum(rows 8-15) on lanes 16-31; one
    // shfl_xor(16) completes the wave total. EXEC is all-1s: no divergence here.
    v2f a; a[0] = acc;  a[1] = 0.0f;
    v2f b; b[0] = 1.0f; b[1] = 1.0f;
    v8f c = {};
    c = __builtin_amdgcn_wmma_f32_16x16x4_f32(
        /*neg_a=*/false, a, /*neg_b=*/false, b,
        /*c_mod=*/(short)0, c, /*reuse_a=*/false, /*reuse_b=*/false);
    float wsum = ((c[0] + c[1]) + (c[2] + c[3])) + ((c[4] + c[5]) + (c[6] + c[7]));
    wsum += __shfl_xor(wsum, 16, 32);

    __shared__ float sw[8];            // 256 threads = 8 wave32
    const int lane = threadIdx.x & 31;
    const int wave = threadIdx.x >> 5;
    if (lane == 0) sw[wave] = wsum;
    __syncthreads();
    if (threadIdx.x == 0) {
        float bsum = 0.0f;
        #pragma unroll
        for (int i = 0; i < 8; ++i) bsum += sw[i];
        partials[blockIdx.x] = bsum;
    }
}

// Kernel 2: deterministic final reduction of block partials (double accum),
// scale by 1/(9N), emit the fp32 scalar.
__global__ void __launch_bounds__(256) nnloss_final(
    const float* __restrict__ partials, int nblocks,
    float* __restrict__ out, double scale)
{
    __shared__ double sd[256];
    double acc = 0.0;
    for (int i = threadIdx.x; i < nblocks; i += 256)
        acc += (double)partials[i];
    sd[threadIdx.x] = acc;
    __syncthreads();
    for (int off = 128; off > 0; off >>= 1) {
        if (threadIdx.x < off) sd[threadIdx.x] += sd[threadIdx.x + off];
        __syncthreads();
    }
    if (threadIdx.x == 0) out[0] = (float)(sd[0] * scale);
}

extern "C" void kernel_launch(void* const* d_in, const int* in_sizes, int n_in,
                              void* d_out, int out_size, void* d_ws, size_t ws_size,
                              hipStream_t stream) {
    const float* fake = (const float*)d_in[0];
    const float* tgt  = (const float*)d_in[1];
    float* out        = (float*)d_out;
    float* partials   = (float*)d_ws;

    const int n     = in_sizes[0];       // B*C*H*W (divisible by 4)
    const int nvec4 = n >> 2;

    int threads = 256;
    int blocks  = (nvec4 + threads - 1) / threads;
    if (blocks > 4096) blocks = 4096;    // grid-stride covers the rest; 16 KB of d_ws
    if (blocks < 1)    blocks = 1;

    nnloss_partial<<<blocks, threads, 0, stream>>>(fake, tgt, partials, nvec4);

    // out = total / (B*H*W) / (C*9) = total / (9*N)
    const double scale = 1.0 / (9.0 * (double)n);
    nnloss_final<<<1, 256, 0, stream>>>(partials, blocks, out, scale);
}